// LLMStyleAttention_65403761984049
// MI455X (gfx1250) — compile-verified
//
#include <hip/hip_runtime.h>
#include <hip/hip_bf16.h>
#include <math.h>

// Problem constants (match reference)
constexpr int cB = 2, cS = 1024, cD = 1024, cH = 16, cHD = 64;
constexpr float cSCALE_T = 0.125f / 0.8f;   // (1/sqrt(64)) / TEMP
constexpr float cTOPP = 0.9f;

typedef __attribute__((ext_vector_type(2))) float v2f;
typedef __attribute__((ext_vector_type(8))) float v8f;

static __device__ __forceinline__ v8f wmma_f32_k4(v2f a, v2f b, v8f c) {
  // D = A(16x4 f32) * B(4x16 f32) + C, wave32, VOP3P
  return __builtin_amdgcn_wmma_f32_16x16x4_f32(false, a, false, b, (short)0, c,
                                               false, false);
}

// ---------------------------------------------------------------------------
// Generic 32x32 output tile GEMM: C[m0..+32, n0..+32] = alpha * A @ B (+bias)
//   A is [M,K] row-major with leading dim lda.
//   BT=true : B is [N,K] row-major (ldb strides N rows)  -> C = A @ B^T
//   BT=false: B is [K,N] row-major (ldb strides K rows)  -> C = A @ B
// Block = 128 threads (4 waves). Wave w computes 16x16 tile (w>>1, w&1).
// K consumed in chunks of 16 staged through LDS; 4 x wmma(16x16x4) per chunk.
// ---------------------------------------------------------------------------
template <bool BT>
__device__ __forceinline__ void gemm_tile32(const float* __restrict__ Ab, int lda,
                                            const float* __restrict__ Bb, int ldb,
                                            float* __restrict__ Cb, int ldc, int K,
                                            float alpha, const float* __restrict__ bias,
                                            int m0, int n0) {
  __shared__ float sA[32][17];  // [m][k], padded to dodge bank conflicts
  __shared__ float sB[32][17];  // always stored [n][k]

  const int t = threadIdx.x;     // 0..127
  const int lane = t & 31;
  const int wave = t >> 5;       // 0..3
  const int wm = (wave >> 1) << 4;
  const int wn = (wave & 1) << 4;
  const int lr = lane & 15;          // row-in-tile (M for A frag, N for B frag)
  const int kh = (lane >> 4) << 1;   // lane-half selects K pair {0,1} vs {2,3}

  v8f c = {0.f, 0.f, 0.f, 0.f, 0.f, 0.f, 0.f, 0.f};

  for (int k0 = 0; k0 < K; k0 += 16) {
    { // A chunk: 32 rows x 16 k, float4-coalesced
      const int row = t >> 2, c4 = (t & 3) << 2;
      const float4 av =
          *(const float4*)(Ab + (size_t)(m0 + row) * lda + k0 + c4);
      sA[row][c4 + 0] = av.x; sA[row][c4 + 1] = av.y;
      sA[row][c4 + 2] = av.z; sA[row][c4 + 3] = av.w;
      if (k0 + 16 < K)
        __builtin_prefetch(Ab + (size_t)(m0 + row) * lda + k0 + 16 + c4, 0, 0);
    }
    if (BT) { // B chunk: 32 n-rows x 16 k
      const int row = t >> 2, c4 = (t & 3) << 2;
      const float4 bv =
          *(const float4*)(Bb + (size_t)(n0 + row) * ldb + k0 + c4);
      sB[row][c4 + 0] = bv.x; sB[row][c4 + 1] = bv.y;
      sB[row][c4 + 2] = bv.z; sB[row][c4 + 3] = bv.w;
    } else {  // B chunk: 16 k-rows x 32 n, transposed into sB[n][k]
      const int kl = t >> 3, c4 = (t & 7) << 2;
      const float4 bv =
          *(const float4*)(Bb + (size_t)(k0 + kl) * ldb + n0 + c4);
      sB[c4 + 0][kl] = bv.x; sB[c4 + 1][kl] = bv.y;
      sB[c4 + 2][kl] = bv.z; sB[c4 + 3][kl] = bv.w;
    }
    __syncthreads();
#pragma unroll
    for (int k4 = 0; k4 < 4; ++k4) {
      const int kb = (k4 << 2) + kh;
      v2f a, b;
      a.x = sA[wm + lr][kb]; a.y = sA[wm + lr][kb + 1];
      b.x = sB[wn + lr][kb]; b.y = sB[wn + lr][kb + 1];
      c = wmma_f32_k4(a, b, c);
    }
    __syncthreads();
  }

  // C/D 16x16 f32 layout: lane%16 = N, vgpr r = M row (lanes>=16 -> M=r+8)
  const int cn = n0 + wn + lr;
  const float bv = bias ? bias[cn] : 0.0f;
  const int mh = (lane >> 4) << 3;
#pragma unroll
  for (int r = 0; r < 8; ++r) {
    const int cm = m0 + wm + r + mh;
    Cb[(size_t)cm * ldc + cn] = c[r] * alpha + bv;
  }
}

// --------------------------- kernels ---------------------------------------

// C[M,N] = A[M,K] @ W[N,K]^T + bias   (projections, output projection)
__global__ void k_gemm_nt(const float* __restrict__ A, const float* __restrict__ W,
                          const float* __restrict__ bias, float* __restrict__ C,
                          int K, int N) {
  gemm_tile32<true>(A, K, W, K, C, N, K, 1.0f, bias,
                    (int)blockIdx.x * 32, (int)blockIdx.y * 32);
}

// scores[b,h,m,n] = (SCALE/TEMP) * sum_d Q[b,m,h*HD+d] * K[b,n,h*HD+d]
__global__ void k_scores(const float* __restrict__ Qp, const float* __restrict__ Kp,
                         float* __restrict__ Sc) {
  const int z = blockIdx.z, b = z / cH, h = z % cH;
  const float* Ab = Qp + (size_t)b * cS * cD + h * cHD;
  const float* Bb = Kp + (size_t)b * cS * cD + h * cHD;
  float* Cb = Sc + (size_t)z * cS * cS;
  gemm_tile32<true>(Ab, cD, Bb, cD, Cb, cS, cHD, cSCALE_T, nullptr,
                    (int)blockIdx.x * 32, (int)blockIdx.y * 32);
}

// Ctx[b,m,h*HD+n] = sum_k attn[b,h,m,k] * V[b,k,h*HD+n]
__global__ void k_attn_v(const float* __restrict__ Attn, const float* __restrict__ Vp,
                         float* __restrict__ Ctx) {
  const int z = blockIdx.z, b = z / cH, h = z % cH;
  const float* Ab = Attn + (size_t)z * cS * cS;
  const float* Bb = Vp + (size_t)b * cS * cD + h * cHD;
  float* Cb = Ctx + (size_t)b * cS * cD + h * cHD;
  gemm_tile32<false>(Ab, cS, Bb, cD, Cb, cD, cS, 1.0f, nullptr,
                     (int)blockIdx.x * 32, (int)blockIdx.y * 32);
}

// Top-p filtering + softmax, one 1024-thread block per score row (in place).
__global__ void k_topp(float* __restrict__ Sc) {
  __shared__ float s_score[cS];
  __shared__ int s_idx[cS];
  __shared__ float s_cum[cS];
  __shared__ int sh_count;
  __shared__ float sh_nuc;

  const int t = threadIdx.x;
  float* p = Sc + (size_t)blockIdx.x * cS;
  const float orig = p[t];

  s_score[t] = orig;
  s_idx[t] = t;
  __syncthreads();

  // Stable bitonic sort: descending score, ties broken by ascending index
  for (int k = 2; k <= cS; k <<= 1) {
    for (int j = k >> 1; j > 0; j >>= 1) {
      const int ixj = t ^ j;
      if (ixj > t) {
        const float a = s_score[t], b = s_score[ixj];
        const int ia = s_idx[t], ib = s_idx[ixj];
        const bool ab = (a > b) || (a == b && ia < ib);  // a precedes b (desc)
        const bool descSeg = ((t & k) == 0);
        if (descSeg ? !ab : ab) {
          s_score[t] = b; s_score[ixj] = a;
          s_idx[t] = ib;  s_idx[ixj] = ia;
        }
      }
      __syncthreads();
    }
  }

  const float maxv = s_score[0];
  // softmax over sorted scores -> inclusive cumsum (Hillis-Steele)
  s_cum[t] = expf(s_score[t] - maxv);
  __syncthreads();
  for (int off = 1; off < cS; off <<= 1) {
    const float add = (t >= off) ? s_cum[t - off] : 0.0f;
    __syncthreads();
    s_cum[t] += add;
    __syncthreads();
  }
  const float total = s_cum[cS - 1];

  // count = 1 + #{j in [0,1022] : cum_j <= TOP_P}   (shifted mask, first=True)
  if (t == 0) sh_count = 1;
  __syncthreads();
  if (t < cS - 1 && (s_cum[t] / total) <= cTOPP) atomicAdd(&sh_count, 1);
  __syncthreads();

  if (t == 0) {
    // replicate reference quirk: cutoff_idx is an ORIGINAL index used to
    // index the SORTED scores
    const int cidx = s_idx[sh_count - 1];
    sh_nuc = s_score[cidx];
  }
  __syncthreads();

  // masked softmax over original scores
  const float nuc = sh_nuc;
  const float e = (orig < nuc) ? 0.0f : expf(orig - maxv);
  s_cum[t] = e;
  __syncthreads();
  for (int s = cS / 2; s > 0; s >>= 1) {
    if (t < s) s_cum[t] += s_cum[t + s];
    __syncthreads();
  }
  p[t] = e / s_cum[0];
}

// --------------------------- launcher ---------------------------------------

extern "C" void kernel_launch(void* const* d_in, const int* in_sizes, int n_in,
                              void* d_out, int out_size, void* d_ws, size_t ws_size,
                              hipStream_t stream) {
  const float* x  = (const float*)d_in[0];
  const float* Wq = (const float*)d_in[1];
  const float* bq = (const float*)d_in[2];
  const float* Wk = (const float*)d_in[3];
  const float* bk = (const float*)d_in[4];
  const float* Wv = (const float*)d_in[5];
  const float* bv = (const float*)d_in[6];
  const float* Wo = (const float*)d_in[7];
  const float* bo = (const float*)d_in[8];

  float* out  = (float*)d_out;                       // [B,S,D]
  float* attn = out + (size_t)cB * cS * cD;          // [B,H,S,S] (scores in place)

  const size_t projN = (size_t)cB * cS * cD;         // 2M floats each
  float* Qp  = (float*)d_ws;
  float* Kp  = Qp + projN;
  float* Vp  = Kp + projN;
  float* Ctx = Vp + projN;
  (void)in_sizes; (void)n_in; (void)out_size; (void)ws_size;

  const dim3 blk(128);
  const dim3 gProj(cB * cS / 32, cD / 32);           // 64 x 32
  k_gemm_nt<<<gProj, blk, 0, stream>>>(x, Wq, bq, Qp, cD, cD);
  k_gemm_nt<<<gProj, blk, 0, stream>>>(x, Wk, bk, Kp, cD, cD);
  k_gemm_nt<<<gProj, blk, 0, stream>>>(x, Wv, bv, Vp, cD, cD);

  const dim3 gSc(cS / 32, cS / 32, cB * cH);         // 32 x 32 x 32
  k_scores<<<gSc, blk, 0, stream>>>(Qp, Kp, attn);

  k_topp<<<cB * cH * cS, cS, 0, stream>>>(attn);     // 32768 blocks x 1024 thr

  const dim3 gAV(cS / 32, cHD / 32, cB * cH);        // 32 x 2 x 32
  k_attn_v<<<gAV, blk, 0, stream>>>(attn, Vp, Ctx);

  k_gemm_nt<<<gProj, blk, 0, stream>>>(Ctx, Wo, bo, out, cD, cD);
}